// YOLOLoss_30399778521440
// MI455X (gfx1250) — compile-verified
//
#include <hip/hip_runtime.h>
#include <math.h>

// ---------------- problem constants ----------------
#define BATCH      64
#define NBOX       50
#define NBOXES     (BATCH * NBOX)          // 3200
#define NCLS       80
#define C5         (5 + NCLS)              // 85
#define M0C        519168                  // 64*52*52*3 cells in pred0
#define M1C        129792                  // 64*26*26*3
#define M2C        32448                   // 64*13*13*3
#define MTOT       (M0C + M1C + M2C)       // 681408
#define NB2        2663                    // ceil(MTOT/256)+1 blocks: 1 cell per thread
#define NB2PAD     2720                    // padded partials region (32-mult)

typedef __attribute__((ext_vector_type(2))) float v2f;
typedef __attribute__((ext_vector_type(8))) float v8f;

__device__ __forceinline__ float softplusf(float x) {
    // bce(x,0) = max(x,0) + log1p(exp(-|x|))
    return fmaxf(x, 0.0f) + log1pf(expf(-fabsf(x)));
}

__device__ __forceinline__ const float* cell_ptr(int cell,
                                                 const float* __restrict__ p0,
                                                 const float* __restrict__ p1,
                                                 const float* __restrict__ p2) {
    if (cell < M0C)        return p0 + (size_t)cell * C5;
    if (cell < M0C + M1C)  return p1 + (size_t)(cell - M0C) * C5;
    return p2 + (size_t)(cell - M0C - M1C) * C5;
}

// XOR-add across lanes with ds_swizzle (group-of-32: src = lane ^ mask).
// Avoids the clamp/cndmask cruft that __shfl_xor generates.
__device__ __forceinline__ float xor_add_fold16(float v) {
    #define SWZ(mask) __int_as_float(__builtin_amdgcn_ds_swizzle( \
        __float_as_int(v), ((mask) << 10) | 0x1F))
    v += SWZ(8);
    v += SWZ(4);
    v += SWZ(2);
    v += SWZ(1);
    #undef SWZ
    return v;
}

// ---------------- kernel 1: per-box cell assignment ----------------
__global__ void box_cell_kernel(const float* __restrict__ bboxes,   // (B,N,4)
                                const float* __restrict__ anchors,  // (9,2)
                                int* __restrict__ cells) {
    int i = blockIdx.x * blockDim.x + threadIdx.x;
    if (i >= NBOXES) return;
    int b = i / NBOX;
    float x = bboxes[4 * i + 0], y = bboxes[4 * i + 1];
    float w = bboxes[4 * i + 2], h = bboxes[4 * i + 3];
    float cx = x + 0.5f * w, cy = y + 0.5f * h;

    // first-occurrence argmin over 9 anchors of L1 diff (matches jnp.argmin)
    int best = 0; float bd = 3.402823e38f;
    #pragma unroll
    for (int j = 0; j < 9; ++j) {
        float d = fabsf(w - anchors[2 * j]) + fabsf(h - anchors[2 * j + 1]);
        if (d < bd) { bd = d; best = j; }
    }
    int s = best / 3, a = best % 3;
    int G    = (s == 0) ? 52 : ((s == 1) ? 26 : 13);
    int base = (s == 0) ? 0  : ((s == 1) ? M0C : (M0C + M1C));
    int ci = (int)floorf(cx * (float)G);   // x index (3rd tgt dim)
    int cj = (int)floorf(cy * (float)G);   // y index (2nd tgt dim)
    ci = min(max(ci, 0), G - 1);
    cj = min(max(cj, 0), G - 1);
    cells[i] = base + ((b * G + cj) * G + ci) * 3 + a;
}

// ---------------- kernel 2: sum softplus over all conf logits -------------
// One thread per cell: every conf load is independent (max MLP; each load is
// a distinct 128B line -> ~87 MB effective traffic, HBM-bound). NT hint keeps
// the 231MB stream from churning the 192MB L2. Wave reduction done with
// v_wmma_f32_16x16x4_f32 (A = ones => column n of D = s_n + s_{n+16}),
// then 4 ds_swizzle XOR folds over the 16 replicated column sums.
__global__ void neg_conf_kernel(const float* __restrict__ p0,
                                const float* __restrict__ p1,
                                const float* __restrict__ p2,
                                float* __restrict__ partials) {
    int c = blockIdx.x * blockDim.x + threadIdx.x;
    float acc = 0.0f;
    if (c < MTOT) {
        const float* q = cell_ptr(c, p0, p1, p2);
        float x = __builtin_nontemporal_load(q + 4);
        acc = softplusf(x);
    }
    // convergent wave reduction via WMMA (EXEC all-1s: full blocks, no early
    // exit; the guarded region above has reconverged)
    v2f a; a[0] = 1.0f; a[1] = 1.0f;           // A = ones
    v2f b; b[0] = acc;  b[1] = 0.0f;           // lane partial into K0/K2 slot
    v8f cacc = {0.f, 0.f, 0.f, 0.f, 0.f, 0.f, 0.f, 0.f};
    v8f d = __builtin_amdgcn_wmma_f32_16x16x4_f32(
        /*neg_a=*/false, a, /*neg_b=*/false, b,
        /*c_mod=*/(short)0, cacc, /*reuse_a=*/false, /*reuse_b=*/false);
    float v = xor_add_fold16(d[0]);            // every lane: wave total

    __shared__ float wsum[8];
    int lane = threadIdx.x & 31, wid = threadIdx.x >> 5;
    if (lane == 0) wsum[wid] = v;
    __syncthreads();
    if (threadIdx.x == 0) {
        float s = 0.0f;
        #pragma unroll
        for (int w2 = 0; w2 < 8; ++w2) s += wsum[w2];   // fixed order
        partials[blockIdx.x] = s;
    }
}

// ---------------- kernel 3: positive-cell terms, dedup by min box idx -----
__global__ void pos_kernel(const float* __restrict__ p0,
                           const float* __restrict__ p1,
                           const float* __restrict__ p2,
                           const int* __restrict__ labels,
                           const int* __restrict__ cells,
                           float* __restrict__ contrib) {  // (NBOXES,4)
    int i = blockIdx.x * blockDim.x + threadIdx.x;
    if (i >= NBOXES) return;
    int b = i / NBOX;
    int cell = cells[i];
    // representative = smallest box index in this batch hitting the same cell
    // (cell index embeds batch, so collisions are intra-batch only)
    bool rep = true;
    for (int j = b * NBOX; j < i; ++j)
        if (cells[j] == cell) { rep = false; break; }

    float pos = 0.f, negsub = 0.f, cls = 0.f, cnt = 0.f;
    if (rep) {
        const float* q = cell_ptr(cell, p0, p1, p2);
        float x4  = q[4];
        float sp4 = softplusf(x4);
        pos    = sp4 - x4;     // bce(x,1)
        negsub = sp4;          // remove this cell from the negative pool
        float csum = 0.f;
        #pragma unroll 4
        for (int c2 = 0; c2 < NCLS; ++c2) csum += softplusf(q[5 + c2]);
        csum -= q[5 + labels[i]];   // bce(x,1)-bce(x,0) = -x on label channel
        cls = csum;
        cnt = 1.f;
    }
    contrib[4 * i + 0] = pos;
    contrib[4 * i + 1] = negsub;
    contrib[4 * i + 2] = cls;
    contrib[4 * i + 3] = cnt;
}

// ---------------- kernel 4: deterministic finalize ----------------
__global__ void finalize_kernel(const float* __restrict__ partials,
                                const float* __restrict__ contrib,
                                float* __restrict__ out) {
    __shared__ float r0[256], r1[256], r2[256], r3[256], r4[256];
    int t = threadIdx.x;
    float s0 = 0.f, s1 = 0.f, s2 = 0.f, s3 = 0.f, sn = 0.f;
    for (int i = t; i < NB2; i += 256) sn += partials[i];
    for (int i = t; i < NBOXES; i += 256) {
        s0 += contrib[4 * i + 0];
        s1 += contrib[4 * i + 1];
        s2 += contrib[4 * i + 2];
        s3 += contrib[4 * i + 3];
    }
    r0[t] = s0; r1[t] = s1; r2[t] = s2; r3[t] = s3; r4[t] = sn;
    __syncthreads();
    for (int off = 128; off > 0; off >>= 1) {
        if (t < off) {
            r0[t] += r0[t + off]; r1[t] += r1[t + off];
            r2[t] += r2[t + off]; r3[t] += r3[t + off];
            r4[t] += r4[t + off];
        }
        __syncthreads();
    }
    if (t == 0) {
        float pos = r0[0], negsub = r1[0], cls = r2[0];
        float npos = r3[0], negtot = r4[0];
        float nposf = fmaxf(npos, 1.0f);
        float nnegf = fmaxf((float)MTOT - npos, 1.0f);
        float conf_loss  = pos / nposf + 0.5f * (negtot - negsub) / nnegf;
        float class_loss = 0.5f * cls / (nposf * (float)NCLS);
        out[0] = conf_loss + class_loss;
    }
}

// ---------------- launcher ----------------
extern "C" void kernel_launch(void* const* d_in, const int* in_sizes, int n_in,
                              void* d_out, int out_size, void* d_ws, size_t ws_size,
                              hipStream_t stream) {
    const float* p0      = (const float*)d_in[0];
    const float* p1      = (const float*)d_in[1];
    const float* p2      = (const float*)d_in[2];
    const float* bboxes  = (const float*)d_in[3];
    const int*   labels  = (const int*)  d_in[4];
    const float* anchors = (const float*)d_in[5];

    float* wsf      = (float*)d_ws;
    float* partials = wsf;                       // NB2 floats (NB2PAD region)
    float* contrib  = wsf + NB2PAD;              // NBOXES*4 floats
    int*   cells    = (int*)(wsf + NB2PAD + NBOXES * 4);  // NBOXES ints
    (void)in_sizes; (void)n_in; (void)out_size; (void)ws_size;

    box_cell_kernel<<<(NBOXES + 255) / 256, 256, 0, stream>>>(bboxes, anchors, cells);
    neg_conf_kernel<<<NB2, 256, 0, stream>>>(p0, p1, p2, partials);
    pos_kernel<<<(NBOXES + 127) / 128, 128, 0, stream>>>(p0, p1, p2, labels, cells, contrib);
    finalize_kernel<<<1, 256, 0, stream>>>(partials, contrib, (float*)d_out);
}